// EEGModel_996432413152
// MI455X (gfx1250) — compile-verified
//
#include <hip/hip_runtime.h>
#include <hip/hip_bf16.h>
#include <math.h>

typedef __attribute__((ext_vector_type(2))) float v2f;
typedef __attribute__((ext_vector_type(8))) float v8f;

#define DEV __device__ __forceinline__

DEV float sigmoidf_(float x){ return 1.0f/(1.0f+__expf(-x)); }

// fp32 WMMA: D(16x16) = A(16x4) * B(4x16) + C. Per-lane operand layout (wave32):
//   half = lane>=16, mn = lane&15
//   A: a.x = A[mn][k+2*half],  a.y = A[mn][k+2*half+1]
//   B: b.x = B[k+2*half][mn],  b.y = B[k+2*half+1][mn]
//   D: acc[r] = D[r + 8*half][mn]
DEV v8f wmma4(v2f a, v2f b, v8f c){
  return __builtin_amdgcn_wmma_f32_16x16x4_f32(false, a, false, b, (short)0, c, false, false);
}

// ---------------- 1) softmax(spectral_w) rows of 128 ----------------
__global__ void k_spectral_softmax(const float* __restrict__ w, float* __restrict__ sm){
  int row = blockIdx.x;      // 64
  int tid = threadIdx.x;     // 128
  __shared__ float red[128];
  float v = w[row*128 + tid];
  red[tid] = v; __syncthreads();
  for(int s=64;s>0;s>>=1){ if(tid<s) red[tid]=fmaxf(red[tid],red[tid+s]); __syncthreads(); }
  float m = red[0]; __syncthreads();
  float e = __expf(v-m);
  red[tid]=e; __syncthreads();
  for(int s=64;s>0;s>>=1){ if(tid<s) red[tid]+=red[tid+s]; __syncthreads(); }
  sm[row*128+tid] = e/red[0];
}

// ---------------- 2) Gram A[b] = (1/T) sum_t Hn_t Hn_t^T  (M=N=64, K=T*F) ----------------
// Fuses spectral scaling (per-thread register cache of sm) and row-normalization (Hn)
// into the LDS staging step; Hn never touches HBM.
__global__ void __launch_bounds__(512) k_gram(const float* __restrict__ x,
                                              const float* __restrict__ sm,
                                              float* __restrict__ A){
  __shared__ float sh[64*128];
  __shared__ float part[64*8];
  __shared__ float ninv[64];
  int b = blockIdx.x;
  int tid = threadIdx.x;                // 512 threads, 16 waves
  int lane = tid & 31, wave = tid >> 5;
  int half = (lane>>4)&1, mn = lane & 15;
  int mt = wave >> 2, nt = wave & 3;    // 4x4 tiles of 16x16 over 64x64
  const float* xb = x + (long long)b*256*8192;
  // each thread's 16 staging slots hit fixed (c,f) positions -> cache sm in regs
  float smv[16];
  #pragma unroll
  for(int j=0;j<16;j++) smv[j] = sm[tid + j*512];
  v8f acc = {0.f,0.f,0.f,0.f,0.f,0.f,0.f,0.f};
  for(int t=0;t<256;t++){
    const float* xt = xb + t*8192;
    #pragma unroll
    for(int j=0;j<16;j++) sh[tid + j*512] = xt[tid + j*512] * smv[j];
    // prefetch next tile: one lane per 128B line covers the full 32KB slice
    if(t < 255 && (tid & 31) == 0){
      const float* xn = xt + 8192;
      #pragma unroll
      for(int j=0;j<16;j++) __builtin_prefetch(xn + tid + j*512, 0, 3);
    }
    __syncthreads();
    { // per-channel L2 norm over F=128
      int row = tid>>3, p = tid&7;
      const float* r = sh + row*128 + p*16;
      float s=0.f;
      #pragma unroll
      for(int j=0;j<16;j++){ float v=r[j]; s += v*v; }
      part[row*8+p]=s;
    }
    __syncthreads();
    if(tid<64){
      float s=0.f;
      #pragma unroll
      for(int j=0;j<8;j++) s+=part[tid*8+j];
      ninv[tid] = 1.0f/(sqrtf(s)+1e-8f);
    }
    __syncthreads();
    { // normalize in place (this is Hn_t)
      int row = tid>>3, p = tid&7;
      float inv = ninv[row];
      float* r = sh + row*128 + p*16;
      #pragma unroll
      for(int j=0;j<16;j++) r[j] *= inv;
    }
    __syncthreads();
    const float* arow = sh + (mt*16+mn)*128;   // A operand: rows of Hn
    const float* brow = sh + (nt*16+mn)*128;   // B operand: B[k][n] = Hn[n][k]
    #pragma unroll 4
    for(int k=0;k<128;k+=4){
      int kk = k + 2*half;
      v2f a;  a.x = arow[kk];  a.y = arow[kk+1];
      v2f bb; bb.x = brow[kk]; bb.y = brow[kk+1];
      acc = wmma4(a, bb, acc);
    }
    __syncthreads();
  }
  float* Ab = A + b*64*64;
  #pragma unroll
  for(int r=0;r<8;r++){
    int m = mt*16 + r + 8*half;
    Ab[m*64 + nt*16 + mn] = acc[r]*(1.0f/256.0f);
  }
}

// ---------------- 3a) top-(K+1)=4 per row ----------------
__global__ void k_topk(const float* __restrict__ A, int* __restrict__ mask){
  int i = blockIdx.x*blockDim.x + threadIdx.x;   // 16*64 rows
  if(i >= 16*64) return;
  const float* row = A + i*64;
  int* mrow = mask + i*64;
  for(int j=0;j<64;j++) mrow[j]=0;
  unsigned long long used = 0ull;
  for(int k=0;k<4;k++){
    float best=-1e30f; int bi=0;
    for(int j=0;j<64;j++){
      if(used & (1ull<<j)) continue;
      float v=row[j];
      if(v>best){best=v;bi=j;}
    }
    used |= (1ull<<bi);
    mrow[bi]=1;
  }
}

// ---------------- 3b) amask = ((m|m^T)&~eye & A>0) | eye ----------------
__global__ void k_amask(const float* __restrict__ A, const int* __restrict__ mask,
                        int* __restrict__ amask){
  int i = blockIdx.x*blockDim.x + threadIdx.x;  // 16*64*64
  if(i >= 16*64*64) return;
  int d = i & 63, c = (i>>6)&63, b = i>>12;
  int m = mask[i] | mask[((b*64+d)<<6)+c];
  int diag = (c==d);
  int mm = m && !diag;
  amask[i] = ((mm && (A[i] > 0.f)) || diag) ? 1 : 0;
}

// ---------------- 4) Wh = (x*sm) @ gat_W   (M=B*T*C, N=32, K=128) ----------------
// Spectral scale fused via LDS-staged sm plane (no xs buffer in HBM).
__global__ void __launch_bounds__(256) k_wh(const float* __restrict__ x,
                                            const float* __restrict__ sm,
                                            const float* __restrict__ W,
                                            float* __restrict__ Wh){
  __shared__ float sW[128*32];
  __shared__ float sSm[64*128];
  long long bt = blockIdx.x;            // 4096 blocks: one (b,t), 64 rows
  int tid = threadIdx.x;                // 256 threads, 8 waves
  for(int j=tid;j<128*32;j+=256) sW[j]=W[j];
  for(int j=tid;j<64*128;j+=256) sSm[j]=sm[j];
  __syncthreads();
  int lane=tid&31, wave=tid>>5;
  int half=(lane>>4)&1, mn=lane&15;
  int mt=wave>>1, nt=wave&1;            // 4 m-tiles x 2 n-tiles
  int c = mt*16 + mn;                   // channel of this lane's A row
  const float* xrow  = x + (bt*64 + c)*128;
  const float* smrow = sSm + c*128;
  v8f acc = {0.f,0.f,0.f,0.f,0.f,0.f,0.f,0.f};
  for(int k=0;k<128;k+=4){
    int kk = k + 2*half;
    v2f a;  a.x = xrow[kk]*smrow[kk]; a.y = xrow[kk+1]*smrow[kk+1];
    v2f bb; bb.x = sW[kk*32 + nt*16+mn]; bb.y = sW[(kk+1)*32 + nt*16+mn];
    acc = wmma4(a, bb, acc);
  }
  float* out = Wh + (bt*64 + mt*16)*32;
  #pragma unroll
  for(int r=0;r<8;r++){
    int m = r + 8*half;
    out[m*32 + nt*16 + mn] = acc[r];
  }
}

// ---------------- 5) GAT attention + pooling, one block per (b,t) ----------------
// s1/s2 computed in-block from the LDS Wh tile (k_s12 kernel + buffers eliminated).
__global__ void __launch_bounds__(256) k_gat(const float* __restrict__ Wh,
                      const float* __restrict__ ga,
                      const int* __restrict__ amask,
                      const float* __restrict__ pw, const float* __restrict__ pb,
                      float* __restrict__ hout){
  __shared__ float sWh[64*32];
  __shared__ float sE[64*64];
  __shared__ float sS1[64], sS2[64], sLog[64], sWpool[64];
  __shared__ float sGat[64*32];
  __shared__ float sRed[2];
  int bt = blockIdx.x;          // b*256+t
  int b = bt >> 8;
  int tid = threadIdx.x;        // 256
  for(int j=tid;j<64*32;j+=256) sWh[j] = Wh[(long long)bt*2048 + j];
  __syncthreads();
  if(tid<64){                    // s1/s2 from LDS tile
    float a=0.f,s=0.f;
    const float* w = sWh + tid*32;
    #pragma unroll
    for(int o=0;o<32;o++){ float v=w[o]; a += v*ga[o]; s += v*ga[32+o]; }
    sS1[tid]=a; sS2[tid]=s;
  }
  __syncthreads();
  const int* am = amask + b*4096;
  for(int j=tid;j<4096;j+=256){
    int c=j>>6, d=j&63;
    float v = sS1[c]+sS2[d];
    v = v>0.f ? v : 0.2f*v;               // leaky_relu 0.2
    sE[j] = am[j] ? v : -1e9f;
  }
  __syncthreads();
  if(tid<64){                              // row softmax over d
    float* row = sE + tid*64;
    float m=-1e30f;
    for(int j=0;j<64;j++) m=fmaxf(m,row[j]);
    float s=0.f;
    for(int j=0;j<64;j++){ float e=__expf(row[j]-m); row[j]=e; s+=e; }
    float inv=1.0f/s;
    for(int j=0;j<64;j++) row[j]*=inv;
  }
  __syncthreads();
  // gat = relu(alpha @ Wh) : M=64, K=64, N=32
  int lane=tid&31, wave=tid>>5;
  int half=(lane>>4)&1, mn=lane&15;
  int mt=wave>>1, nt=wave&1;
  v8f acc={0.f,0.f,0.f,0.f,0.f,0.f,0.f,0.f};
  const float* arow = sE + (mt*16+mn)*64;
  for(int k=0;k<64;k+=4){
    int kk=k+2*half;
    v2f a;  a.x=arow[kk]; a.y=arow[kk+1];
    v2f bb; bb.x=sWh[kk*32 + nt*16+mn]; bb.y=sWh[(kk+1)*32 + nt*16+mn];
    acc = wmma4(a, bb, acc);
  }
  #pragma unroll
  for(int r=0;r<8;r++){
    int m = mt*16 + r + 8*half;
    sGat[m*32 + nt*16+mn] = fmaxf(acc[r], 0.f);
  }
  __syncthreads();
  if(tid<64){                              // pool logits over channels
    float s=pb[0];
    for(int o=0;o<32;o++) s += sGat[tid*32+o]*pw[o];
    sLog[tid]=s;
  }
  __syncthreads();
  if(tid==0){
    float m=-1e30f; for(int j=0;j<64;j++) m=fmaxf(m,sLog[j]);
    float s=0.f;    for(int j=0;j<64;j++) s+=__expf(sLog[j]-m);
    sRed[0]=m; sRed[1]=s;
  }
  __syncthreads();
  if(tid<64) sWpool[tid] = __expf(sLog[tid]-sRed[0])/sRed[1];
  __syncthreads();
  if(tid<32){
    float s=0.f;
    for(int c=0;c<64;c++) s += sGat[c*32+tid]*sWpool[c];
    hout[(long long)bt*32 + tid] = s;
  }
}

// ---------------- 6) one LSTM direction (block = 256 thr, 8 waves) ----------------
__global__ void __launch_bounds__(256) k_lstm(const float* __restrict__ hin,   // (16,256,32)
                       const float* __restrict__ Wi, const float* __restrict__ Whh,
                       const float* __restrict__ bias,
                       float* __restrict__ hcat, int reverse){
  __shared__ float sWi[32*128];
  __shared__ float sWh[32*128];
  __shared__ float sB[128];
  __shared__ float sH[16*32];
  __shared__ float sC[16*32];
  __shared__ float sX[16*32];
  __shared__ float sG[16*128];
  int tid=threadIdx.x;
  for(int j=tid;j<4096;j+=256){ sWi[j]=Wi[j]; sWh[j]=Whh[j]; }
  if(tid<128) sB[tid]=bias[tid];
  for(int j=tid;j<512;j+=256){ sH[j]=0.f; sC[j]=0.f; }
  __syncthreads();
  int lane=tid&31, wave=tid>>5;           // wave = N-tile (8 x 16 cols = 128 gates)
  int half=(lane>>4)&1, mn=lane&15;
  for(int step=0; step<256; step++){
    int t = reverse ? (255-step) : step;
    for(int j=tid;j<512;j+=256){
      int bb=j>>5, o=j&31;
      sX[j] = hin[((long long)bb*256 + t)*32 + o];
    }
    __syncthreads();
    v8f acc;
    float bv = sB[wave*16+mn];
    #pragma unroll
    for(int r=0;r<8;r++) acc[r]=bv;
    for(int k=0;k<32;k+=4){               // x_t @ Wi
      int kk=k+2*half;
      v2f a;  a.x=sX[mn*32+kk]; a.y=sX[mn*32+kk+1];
      v2f bb2; bb2.x=sWi[kk*128 + wave*16+mn]; bb2.y=sWi[(kk+1)*128 + wave*16+mn];
      acc = wmma4(a, bb2, acc);
    }
    for(int k=0;k<32;k+=4){               // h @ Whh
      int kk=k+2*half;
      v2f a;  a.x=sH[mn*32+kk]; a.y=sH[mn*32+kk+1];
      v2f bb2; bb2.x=sWh[kk*128 + wave*16+mn]; bb2.y=sWh[(kk+1)*128 + wave*16+mn];
      acc = wmma4(a, bb2, acc);
    }
    #pragma unroll
    for(int r=0;r<8;r++){
      int m=r+8*half;
      sG[m*128 + wave*16+mn] = acc[r];
    }
    __syncthreads();
    for(int j=tid;j<512;j+=256){          // gates
      int m=j>>5, o=j&31;
      float gi=sG[m*128+o], gf=sG[m*128+32+o], gg=sG[m*128+64+o], go=sG[m*128+96+o];
      float c = sigmoidf_(gf)*sC[j] + sigmoidf_(gi)*tanhf(gg);
      float h = sigmoidf_(go)*tanhf(c);
      sC[j]=c; sH[j]=h;
      hcat[((long long)m*256 + t)*64 + reverse*32 + o] = h;
    }
    __syncthreads();
  }
}

// ---------------- 7) temporal attention + LN + FC, one block per batch ----------------
__global__ void __launch_bounds__(256) k_final(const float* __restrict__ hcat,
                        const float* __restrict__ taW, const float* __restrict__ tab,
                        const float* __restrict__ tav,
                        const float* __restrict__ lng, const float* __restrict__ lnb,
                        const float* __restrict__ f1w, const float* __restrict__ f1b,
                        const float* __restrict__ f2w, const float* __restrict__ f2b,
                        float* __restrict__ out){
  __shared__ float sW[64*64];
  __shared__ float sb[64], sv[64];
  __shared__ float sSc[256];
  __shared__ float sRed[256];
  __shared__ float sCtx[64];
  __shared__ float sR1[32];
  __shared__ float sMV[2];
  int b=blockIdx.x, tid=threadIdx.x;     // 256 threads (t index)
  for(int j=tid;j<4096;j+=256) sW[j]=taW[j];
  if(tid<64){ sb[tid]=tab[tid]; sv[tid]=tav[tid]; }
  __syncthreads();
  const float* hb = hcat + (long long)b*256*64;
  {
    const float* hr = hb + tid*64;
    float hl[64];
    for(int i=0;i<64;i++) hl[i]=hr[i];
    float sc=0.f;
    for(int j=0;j<64;j++){
      float tmp=sb[j];
      for(int i=0;i<64;i++) tmp += hl[i]*sW[i*64+j];
      sc += tanhf(tmp)*sv[j];
    }
    sSc[tid] = sc*(1.0f/1.5f);
  }
  __syncthreads();
  sRed[tid]=sSc[tid]; __syncthreads();
  for(int s=128;s>0;s>>=1){ if(tid<s) sRed[tid]=fmaxf(sRed[tid],sRed[tid+s]); __syncthreads(); }
  float m=sRed[0]; __syncthreads();
  float e=__expf(sSc[tid]-m);
  sRed[tid]=e; __syncthreads();
  for(int s=128;s>0;s>>=1){ if(tid<s) sRed[tid]+=sRed[tid+s]; __syncthreads(); }
  float al = e/sRed[0];
  __syncthreads();
  sSc[tid]=al;
  __syncthreads();
  if(tid<64){
    float s=0.f;
    for(int t=0;t<256;t++) s += hb[t*64+tid]*sSc[t];
    sCtx[tid]=s;
  }
  __syncthreads();
  if(tid==0){
    float mu=0.f; for(int i=0;i<64;i++) mu+=sCtx[i]; mu *= (1.0f/64.0f);
    float va=0.f; for(int i=0;i<64;i++){ float d=sCtx[i]-mu; va+=d*d; } va *= (1.0f/64.0f);
    sMV[0]=mu; sMV[1]=1.0f/sqrtf(va+1e-5f);
  }
  __syncthreads();
  if(tid<64) sCtx[tid] = (sCtx[tid]-sMV[0])*sMV[1]*lng[tid] + lnb[tid];
  __syncthreads();
  if(tid<32){
    float s=f1b[tid];
    for(int i=0;i<64;i++) s += sCtx[i]*f1w[i*32+tid];
    sR1[tid]=fmaxf(s,0.f);
  }
  __syncthreads();
  if(tid<3){
    float s=f2b[tid];
    for(int i=0;i<32;i++) s += sR1[i]*f2w[i*3+tid];
    out[b*3+tid]=s;
  }
}

extern "C" void kernel_launch(void* const* d_in, const int* in_sizes, int n_in,
                              void* d_out, int out_size, void* d_ws, size_t ws_size,
                              hipStream_t stream) {
  (void)in_sizes; (void)n_in; (void)out_size; (void)ws_size;
  const float* x          = (const float*)d_in[0];
  const float* spectral_w = (const float*)d_in[1];
  const float* gat_W      = (const float*)d_in[2];
  const float* gat_a      = (const float*)d_in[3];
  const float* pool_w     = (const float*)d_in[4];
  const float* pool_b     = (const float*)d_in[5];
  const float* lstm_Wi_f  = (const float*)d_in[6];
  const float* lstm_Wh_f  = (const float*)d_in[7];
  const float* lstm_b_f   = (const float*)d_in[8];
  const float* lstm_Wi_r  = (const float*)d_in[9];
  const float* lstm_Wh_r  = (const float*)d_in[10];
  const float* lstm_b_r   = (const float*)d_in[11];
  const float* ta_W       = (const float*)d_in[12];
  const float* ta_b       = (const float*)d_in[13];
  const float* ta_v       = (const float*)d_in[14];
  const float* ln_g       = (const float*)d_in[15];
  const float* ln_b       = (const float*)d_in[16];
  const float* fc1_w      = (const float*)d_in[17];
  const float* fc1_b      = (const float*)d_in[18];
  const float* fc2_w      = (const float*)d_in[19];
  const float* fc2_b      = (const float*)d_in[20];

  float* ws = (float*)d_ws;
  size_t off = 0;
  float* sm    = ws + off; off += 64*128;
  float* A     = ws + off; off += (size_t)16*64*64;
  int*   mask  = (int*)(ws + off); off += (size_t)16*64*64;
  int*   amask = (int*)(ws + off); off += (size_t)16*64*64;
  float* Wh    = ws + off; off += (size_t)16*256*64*32;
  float* hp    = ws + off; off += (size_t)16*256*32;
  float* hc    = ws + off; off += (size_t)16*256*64;

  k_spectral_softmax<<<64,128,0,stream>>>(spectral_w, sm);
  k_gram<<<16,512,0,stream>>>(x, sm, A);
  k_topk<<<4,256,0,stream>>>(A, mask);
  k_amask<<<256,256,0,stream>>>(A, mask, amask);
  k_wh<<<4096,256,0,stream>>>(x, sm, gat_W, Wh);
  k_gat<<<4096,256,0,stream>>>(Wh, gat_a, amask, pool_w, pool_b, hp);
  k_lstm<<<1,256,0,stream>>>(hp, lstm_Wi_f, lstm_Wh_f, lstm_b_f, hc, 0);
  k_lstm<<<1,256,0,stream>>>(hp, lstm_Wi_r, lstm_Wh_r, lstm_b_r, hc, 1);
  k_final<<<16,256,0,stream>>>(hc, ta_W, ta_b, ta_v, ln_g, ln_b,
                               fc1_w, fc1_b, fc2_w, fc2_b, (float*)d_out);
}